// Greedy_33981781246429
// MI455X (gfx1250) — compile-verified
//
#include <hip/hip_runtime.h>
#include <hip/hip_bf16.h>
#include <stdint.h>

#define L 2048
#define LT 128  // 2048/16 tiles per dimension

typedef float v2f __attribute__((ext_vector_type(2)));
typedef float v8f __attribute__((ext_vector_type(8)));

// Allowed-pair bitmask over class indices {0:2, 1:3, 2:5, 3:7}.
// products 6=(0,1),(1,0); 35=(2,3),(3,2); 21=(1,3),(3,1)
// bit k*4+c set for allowed (k,c): bits 1,4,7,11,13,14 -> 26770
#define MMASK 26770u

// ---------------------------------------------------------------------------
// Kernel 1: per-row prime class = argmax over feat[0, 0:4, r, 0] (first max).
// ---------------------------------------------------------------------------
__global__ void k_cls(const float* __restrict__ feat, int* __restrict__ cls) {
  int r = blockIdx.x * blockDim.x + threadIdx.x;
  if (r >= L) return;
  const size_t LL = (size_t)L * (size_t)L;
  float best = feat[(size_t)r * L];
  int bc = 0;
#pragma unroll
  for (int c = 1; c < 4; ++c) {
    float v = feat[(size_t)c * LL + (size_t)r * L];
    if (v > best) { best = v; bc = c; }  // strict > keeps first max (jnp.argmax)
  }
  cls[r] = bc;
}

// ---------------------------------------------------------------------------
// Kernel 2: build c2 = band ⊙ canonical ⊙ (con + conᵀ)/2, stored into d_out.
// Canonical mask per 16x16 tile computed with one V_WMMA_F32_16X16X4_F32:
//   mask_tile = A(16x4 one-hot of row classes) x B(4x16, B[k][n]=M[k][cls_jn]).
// One wave per tile; 8 waves (256 threads) per block; 16384 tiles total.
// ---------------------------------------------------------------------------
__global__ void k_build(const float* __restrict__ con,
                        const int* __restrict__ cls,
                        float* __restrict__ out) {
  const int wave = threadIdx.x >> 5;
  const int lane = threadIdx.x & 31;
  const int tile = blockIdx.x * 8 + wave;
  const int ti = tile >> 7;          // tile row    (0..127)
  const int tj = tile & (LT - 1);    // tile column (0..127)
  const int i0 = ti << 4, j0 = tj << 4;
  const int m = lane & 15;           // N / M-sub index
  const int h = lane >> 4;           // lane half selects K pair {2h, 2h+1}

  const int ci = cls[i0 + m];        // class of matrix row  i0+m  (A operand)
  const int cj = cls[j0 + m];        // class of matrix col  j0+m  (B operand)

  v8f d;
#if __has_builtin(__builtin_amdgcn_wmma_f32_16x16x4_f32)
  // A 16x4 f32 layout (ISA 7.12.2): lanes0-15 VGPR0->K=0,VGPR1->K=1;
  //                                 lanes16-31 VGPR0->K=2,VGPR1->K=3; M=lane%16
  v2f a, b;
  a.x = (ci == 2 * h) ? 1.0f : 0.0f;
  a.y = (ci == 2 * h + 1) ? 1.0f : 0.0f;
  // B 4x16: lane half splits K in halves, VGPRs enumerate K within half; N=lane%16
  b.x = (float)((MMASK >> ((2 * h) * 4 + cj)) & 1u);
  b.y = (float)((MMASK >> ((2 * h + 1) * 4 + cj)) & 1u);
  v8f cacc = {};
  d = __builtin_amdgcn_wmma_f32_16x16x4_f32(false, a, false, b,
                                            (short)0, cacc, false, false);
#else
  // Scalar fallback: mask[i][j] = M[cls_i][cls_j]
#pragma unroll
  for (int v = 0; v < 8; ++v) {
    int crow = cls[i0 + v + 8 * h];
    d[v] = (float)((MMASK >> (crow * 4 + cj)) & 1u);
  }
#endif

  // D layout (documented): lane -> (M = v + 8h, N = m), component v in 0..7
#pragma unroll
  for (int v = 0; v < 8; ++v) {
    const int i = i0 + v + 8 * h;
    const int j = j0 + m;
    int dd = i - j; dd = dd < 0 ? -dd : dd;
    float val = 0.0f;
    if (d[v] > 0.5f && dd >= 4) {
      val = 0.5f * (con[(size_t)i * L + j] + con[(size_t)j * L + i]);
    }
    out[(size_t)i * L + j] = val;
  }
}

// ---------------------------------------------------------------------------
// Kernel 3: persistent single-workgroup locally-dominant (Preis) matching.
// Exactly reproduces the sequential greedy on positive entries.
// 1024 threads = 32 waves; wave w owns rows r ≡ w (mod 32); lanes scan a row
// cooperatively (coalesced) and butterfly-reduce (max value, min index on tie).
// State in LDS; rounds separated by workgroup barriers; provably <=1025 rounds.
// ---------------------------------------------------------------------------
__global__ void __launch_bounds__(1024)
k_match(const float* __restrict__ c2, int* __restrict__ partner) {
  __shared__ int s_idx[L];              // -1 = needs scan, -2 = dead, else best j
  __shared__ unsigned char s_used[L];
  __shared__ int s_mlist[1024];
  __shared__ int s_nmatch;

  const int tid = threadIdx.x;
  const int wave = tid >> 5;
  const int lane = tid & 31;

  for (int r = tid; r < L; r += 1024) {
    s_idx[r] = -1;
    s_used[r] = 0;
    partner[r] = -1;
  }
  __syncthreads();

  for (int round = 0; round < 1100; ++round) {
    if (tid == 0) s_nmatch = 0;
    __syncthreads();

    // --- scan phase: rescan rows whose cached best is invalid ---
    for (int r = wave; r < L; r += 32) {
      if (s_used[r]) continue;
      int bi = s_idx[r];
      if (bi == -2) continue;                 // dead: no positive free partner
      if (bi >= 0 && !s_used[bi]) continue;   // cached best still valid
      const float* row = c2 + (size_t)r * L;
      float bv = 0.0f;
      int bj = -1;
      for (int t = 0; t < L / 32; ++t) {
        int j = lane + 32 * t;
        float v = row[j];
        if (v > 0.0f && !s_used[j]) {
          if (v > bv) { bv = v; bj = j; }     // strict > keeps smallest j on tie
        }
      }
      // butterfly all-reduce: larger value wins; on equal value, smaller index
      for (int off = 16; off > 0; off >>= 1) {
        float ov = __shfl_xor(bv, off);
        int oj = __shfl_xor(bj, off);
        bool take = (oj >= 0) &&
                    (bj < 0 || ov > bv || (ov == bv && oj < bj));
        if (take) { bv = ov; bj = oj; }
      }
      if (lane == 0) s_idx[r] = (bj < 0) ? -2 : bj;
    }
    __syncthreads();

    // --- detect mutual pairs ---
    for (int r = tid; r < L; r += 1024) {
      if (s_used[r]) continue;
      int j = s_idx[r];
      if (j > r && !s_used[j] && s_idx[j] == r) {
        int pos = atomicAdd(&s_nmatch, 1);
        s_mlist[pos] = (r << 16) | j;
      }
    }
    __syncthreads();

    int nm = s_nmatch;
    // --- apply matches ---
    for (int k = tid; k < nm; k += 1024) {
      int i = s_mlist[k] >> 16;
      int j = s_mlist[k] & 0xFFFF;
      s_used[i] = 1; s_used[j] = 1;
      partner[i] = j; partner[j] = i;
    }
    __syncthreads();

    if (nm == 0) break;  // no free-free positive edge remains -> done
  }
}

// ---------------------------------------------------------------------------
// Kernel 4: finalize in place — keep c2[i][j] only where partner[i]==j.
// ---------------------------------------------------------------------------
__global__ void k_final(float* __restrict__ out, const int* __restrict__ partner) {
  size_t idx = (size_t)blockIdx.x * blockDim.x + threadIdx.x;
  int i = (int)(idx >> 11);
  int j = (int)(idx & (L - 1));
  float v = out[idx];
  out[idx] = (partner[i] == j) ? v : 0.0f;
}

// ---------------------------------------------------------------------------
extern "C" void kernel_launch(void* const* d_in, const int* in_sizes, int n_in,
                              void* d_out, int out_size, void* d_ws, size_t ws_size,
                              hipStream_t stream) {
  const float* con = (const float*)d_in[0];   // (1,1,2048,2048) f32
  const float* feat = (const float*)d_in[1];  // (1,8,2048,2048) f32
  float* out = (float*)d_out;                 // (1,1,2048,2048) f32

  int* cls = (int*)d_ws;          // 2048 ints  (8 KB)
  int* partner = cls + L;         // 2048 ints  (8 KB)

  k_cls<<<(L + 255) / 256, 256, 0, stream>>>(feat, cls);
  k_build<<<(LT * LT) / 8, 256, 0, stream>>>(con, cls, out);   // c2 staged in d_out
  k_match<<<1, 1024, 0, stream>>>(out, partner);
  k_final<<<(int)(((size_t)L * L) / 256), 256, 0, stream>>>(out, partner);
}